// StatelessNTMWriteHead_17798344475182
// MI455X (gfx1250) — compile-verified
//
#include <hip/hip_runtime.h>
#include <math.h>
#include <stdint.h>

typedef __attribute__((ext_vector_type(2))) float v2f;
typedef __attribute__((ext_vector_type(4))) float v4f;
typedef __attribute__((ext_vector_type(8))) float v8f;

#define Bq 64
#define Nq 16384
#define Mq 64
#define Cq 512
#define OUTq 198
#define EPSq 1e-8f
#define LSTRIDE 68  // 68*4B row stride: 16B-aligned slots + conflict-free column reads

__device__ __forceinline__ float softplus_f(float x) {
  return (x > 20.f) ? x : log1pf(expf(x));
}
__device__ __forceinline__ float sigmoid_f(float x) {
  return 1.f / (1.f + expf(-x));
}

// ---------------------------------------------------------------------------
// K1: proj = x @ W + b  via V_WMMA_F32_16X16X4_F32
// grid (13,4), block 32 (one wave per 16x16 tile). K = 512 in steps of 4.
// ---------------------------------------------------------------------------
__global__ void k_proj(const float* __restrict__ x, const float* __restrict__ W,
                       const float* __restrict__ bias, float* __restrict__ proj) {
  const int lane = threadIdx.x;
  const int colBase = blockIdx.x * 16;
  const int rowBase = blockIdx.y * 16;
  const int row = rowBase + (lane & 15);
  const int col = colBase + (lane & 15);
  const int kb = (lane >> 4) * 2;
  const int colc = (col < OUTq) ? col : (OUTq - 1);
  const float mask = (col < OUTq) ? 1.f : 0.f;

  v8f c = {};
  for (int k = 0; k < Cq; k += 4) {
    v2f a, bm;
    a.x = x[row * Cq + k + kb];
    a.y = x[row * Cq + k + kb + 1];
    bm.x = W[(k + kb) * OUTq + colc] * mask;
    bm.y = W[(k + kb + 1) * OUTq + colc] * mask;
    c = __builtin_amdgcn_wmma_f32_16x16x4_f32(false, a, false, bm, (short)0, c,
                                              false, false);
  }
  if (col < OUTq) {
    const float bb = bias[col];
    const int rb = rowBase + ((lane >> 4) ? 8 : 0);
#pragma unroll
    for (int r = 0; r < 8; ++r) {
      proj[(rb + r) * OUTq + col] = c[r] + bb;
    }
  }
}

// ---------------------------------------------------------------------------
// K2: per-batch scalars. scal layout per b (stride 16):
// 0 beta, 1 g, 2..4 s, 5 gamma, 6 knorm, 7 maxlogit, 8 Z, 9 S
// ---------------------------------------------------------------------------
__global__ void k_scalars(const float* __restrict__ proj, float* __restrict__ scal,
                          float* __restrict__ e_arr, float* __restrict__ a_arr) {
  __shared__ float red[64];
  const int b = blockIdx.x;
  const int t = threadIdx.x;
  const float* p = proj + b * OUTq;
  const float kv = (t < Mq) ? p[t] : 0.f;
  red[t] = kv * kv;
  __syncthreads();
  for (int s = 32; s > 0; s >>= 1) {
    if (t < s) red[t] += red[t + s];
    __syncthreads();
  }
  if (t == 0) {
    float* sc = scal + b * 16;
    sc[6] = sqrtf(red[0]);                 // ||k||
    sc[0] = softplus_f(p[Mq]);             // beta
    sc[1] = sigmoid_f(p[Mq + 1]);          // g
    const float s0 = p[Mq + 2], s1 = p[Mq + 3], s2 = p[Mq + 4];
    const float mx = fmaxf(s0, fmaxf(s1, s2));
    const float e0 = expf(s0 - mx), e1 = expf(s1 - mx), e2 = expf(s2 - mx);
    const float inv = 1.f / (e0 + e1 + e2);
    sc[2] = e0 * inv; sc[3] = e1 * inv; sc[4] = e2 * inv;
    sc[5] = 1.f + softplus_f(p[Mq + 5]);   // gamma
  }
  if (t < Mq) {
    e_arr[b * Mq + t] = sigmoid_f(p[Mq + 6 + t]);
    a_arr[b * Mq + t] = p[Mq + 6 + Mq + t];
  }
}

// ---------------------------------------------------------------------------
// K3: logits[b,n] = beta * dot(memory[b,n,:], k[b]) / (||row||*||k|| + eps)
// 8 waves/block; each wave stages a 16x64 f32 tile into LDS with
// GLOBAL_LOAD_ASYNC_TO_LDS_B128 (ASYNCcnt), then 16 WMMA f32 16x16x4 steps.
// LDS offset is derived from the real address of `tile` so the array escapes
// into the asm (ptrtoint capture) — otherwise LLVM proves `tile` has no
// stores and folds the fragment loads (and the WMMAs) to poison.
// ---------------------------------------------------------------------------
__global__ void k_logits(const float* __restrict__ mem,
                         const float* __restrict__ proj,
                         const float* __restrict__ scal,
                         float* __restrict__ logits) {
  __shared__ float tile[8][16][LSTRIDE];
  const int lane = threadIdx.x & 31;
  const int wv = threadIdx.x >> 5;
  const int blk = blockIdx.x;
  const int b = blk >> 7;                 // / (N/128)
  const int n0 = ((blk & 127) << 7) + wv * 16;
  const float* rowbase = mem + ((size_t)b * Nq + n0) * Mq;

  // async stage 16 rows x 64 floats (each lane: one 16B chunk per iteration)
#pragma unroll
  for (int it = 0; it < 8; ++it) {
    const int idx = it * 128 + lane * 4;
    const int r = idx >> 6, cm = idx & 63;
    // low 32 bits of the flat LDS-aperture address == LDS byte offset
    const unsigned lds_off = (unsigned)(uintptr_t)(&tile[wv][r][cm]);
    const float* gp = rowbase + r * Mq + cm;
    asm volatile("global_load_async_to_lds_b128 %0, %1, off"
                 :: "v"(lds_off), "v"(gp)
                 : "memory");
  }
  asm volatile("s_wait_asynccnt 0x0" ::: "memory");
  __syncthreads();

  const float* kvec = proj + b * OUTq;  // k = proj[b, 0:M]
  const int rrow = lane & 15;
  const int kb = (lane >> 4) * 2;
  v8f c = {};
  float nrm2 = 0.f;
#pragma unroll
  for (int s = 0; s < 16; ++s) {
    const int kidx = 4 * s + kb;
    v2f a, bm;
    a.x = tile[wv][rrow][kidx];
    a.y = tile[wv][rrow][kidx + 1];
    bm.x = kvec[kidx];
    bm.y = kvec[kidx + 1];
    nrm2 += a.x * a.x + a.y * a.y;
    c = __builtin_amdgcn_wmma_f32_16x16x4_f32(false, a, false, bm, (short)0, c,
                                              false, false);
  }
  nrm2 += __shfl_xor(nrm2, 16, 32);  // combine half-row partials

  // gather per-row dots onto lanes 0..15 (all 16 columns of D are identical)
  float myDot = 0.f;
#pragma unroll
  for (int r = 0; r < 8; ++r) {
    const float hi = __shfl(c[r], 16, 32);  // lanes>=16 hold rows 8..15
    if (lane == r) myDot = c[r];
    if (lane == 8 + r) myDot = hi;
  }
  if (lane < 16) {
    const float* sc = scal + b * 16;
    const float norm = sqrtf(nrm2);
    logits[(size_t)b * Nq + n0 + lane] =
        sc[0] * myDot / (norm * sc[6] + EPSq);
  }
}

// ---------------------------------------------------------------------------
// K4: per-batch softmax stats (max, sum of exp). one block per batch.
// ---------------------------------------------------------------------------
__global__ void k_stats(const float* __restrict__ logits, float* __restrict__ scal) {
  __shared__ float red[256];
  const int b = blockIdx.x;
  const int t = threadIdx.x;
  const float* l = logits + (size_t)b * Nq;
  float mx = -1e30f;
  for (int n = t; n < Nq; n += 256) mx = fmaxf(mx, l[n]);
  red[t] = mx;
  __syncthreads();
  for (int s = 128; s > 0; s >>= 1) {
    if (t < s) red[t] = fmaxf(red[t], red[t + s]);
    __syncthreads();
  }
  const float m = red[0];
  __syncthreads();
  float sum = 0.f;
  for (int n = t; n < Nq; n += 256) sum += expf(l[n] - m);
  red[t] = sum;
  __syncthreads();
  for (int s = 128; s > 0; s >>= 1) {
    if (t < s) red[t] += red[t + s];
    __syncthreads();
  }
  if (t == 0) { scal[b * 16 + 7] = m; scal[b * 16 + 8] = red[0]; }
}

// ---------------------------------------------------------------------------
// K5: w_g = g * softmax(logits) + (1-g) * w_prev
// ---------------------------------------------------------------------------
__global__ void k_wg(const float* __restrict__ logits,
                     const float* __restrict__ w_prev,
                     const float* __restrict__ scal, float* __restrict__ wg) {
  const size_t i = (size_t)blockIdx.x * 256 + threadIdx.x;
  const int b = (int)(i >> 14);  // N = 2^14
  const float* sc = scal + b * 16;
  const float wc = expf(logits[i] - sc[7]) / sc[8];
  const float g = sc[1];
  wg[i] = g * wc + (1.f - g) * w_prev[i];
}

// ---------------------------------------------------------------------------
// K6: circular shift + sharpen, per-batch sum. one block per batch.
// ---------------------------------------------------------------------------
__global__ void k_sharpen(const float* __restrict__ wg, float* __restrict__ scal,
                          float* __restrict__ wsh) {
  __shared__ float red[256];
  const int b = blockIdx.x;
  const int t = threadIdx.x;
  const float* w = wg + (size_t)b * Nq;
  const float* sc = scal + b * 16;
  const float s0 = sc[2], s1 = sc[3], s2 = sc[4], gamma = sc[5];
  float sum = 0.f;
  for (int n = t; n < Nq; n += 256) {
    const float left = w[(n + 1) & (Nq - 1)];   // roll(-1)
    const float right = w[(n - 1) & (Nq - 1)];  // roll(+1)
    const float wsv = s0 * left + s1 * w[n] + s2 * right;
    const float v = powf(wsv + EPSq, gamma);
    wsh[(size_t)b * Nq + n] = v;
    sum += v;
  }
  red[t] = sum;
  __syncthreads();
  for (int s = 128; s > 0; s >>= 1) {
    if (t < s) red[t] += red[t + s];
    __syncthreads();
  }
  if (t == 0) scal[b * 16 + 9] = red[0];
}

// ---------------------------------------------------------------------------
// K7: w = wsh / S; mem_new = memory*(1 - w*e) + w*a
// Streaming 256MB in + 256MB out with zero reuse -> non-temporal hints so
// the 4MB softmax buffers stay L2-resident.
// ---------------------------------------------------------------------------
__global__ void k_update(const float* __restrict__ mem,
                         const float* __restrict__ wsh,
                         const float* __restrict__ scal,
                         const float* __restrict__ e_arr,
                         const float* __restrict__ a_arr,
                         float* __restrict__ out_w,
                         float* __restrict__ out_mem) {
  const size_t i = (size_t)blockIdx.x * 256 + threadIdx.x;
  const int m4 = (int)(i & 15);
  const size_t bn = i >> 4;
  const int b = (int)(bn >> 14);
  const float w = wsh[bn] / scal[b * 16 + 9];
  if (m4 == 0) out_w[bn] = w;
  const int m = m4 * 4;
  const v4f e4 = *(const v4f*)(e_arr + b * Mq + m);
  const v4f a4 = *(const v4f*)(a_arr + b * Mq + m);
  const v4f mv = __builtin_nontemporal_load((const v4f*)(mem + bn * Mq + m));
  v4f o;
  o.x = mv.x * (1.f - w * e4.x) + w * a4.x;
  o.y = mv.y * (1.f - w * e4.y) + w * a4.y;
  o.z = mv.z * (1.f - w * e4.z) + w * a4.z;
  o.w = mv.w * (1.f - w * e4.w) + w * a4.w;
  __builtin_nontemporal_store(o, (v4f*)(out_mem + bn * Mq + m));
}

// ---------------------------------------------------------------------------
extern "C" void kernel_launch(void* const* d_in, const int* in_sizes, int n_in,
                              void* d_out, int out_size, void* d_ws, size_t ws_size,
                              hipStream_t stream) {
  const float* x      = (const float*)d_in[0];  // (B, C)
  const float* w_prev = (const float*)d_in[1];  // (B, N)
  const float* memory = (const float*)d_in[2];  // (B, N, M)
  const float* W      = (const float*)d_in[3];  // (C, OUT)
  const float* bias   = (const float*)d_in[4];  // (OUT,)

  float* proj    = (float*)d_out;               // B*OUT
  float* out_w   = proj + Bq * OUTq;            // B*N
  float* out_mem = out_w + (size_t)Bq * Nq;     // B*N*M

  float* ws     = (float*)d_ws;
  float* logits = ws;                           // B*N
  float* wg     = logits + (size_t)Bq * Nq;     // B*N
  float* wsh    = wg + (size_t)Bq * Nq;         // B*N
  float* scal   = wsh + (size_t)Bq * Nq;        // B*16
  float* e_arr  = scal + Bq * 16;               // B*M
  float* a_arr  = e_arr + Bq * Mq;              // B*M

  k_proj<<<dim3(13, 4), 32, 0, stream>>>(x, W, bias, proj);
  k_scalars<<<Bq, 64, 0, stream>>>(proj, scal, e_arr, a_arr);
  k_logits<<<Bq * (Nq / 128), 256, 0, stream>>>(memory, proj, scal, logits);
  k_stats<<<Bq, 256, 0, stream>>>(logits, scal);
  k_wg<<<(Bq * Nq) / 256, 256, 0, stream>>>(logits, w_prev, scal, wg);
  k_sharpen<<<Bq, 256, 0, stream>>>(wg, scal, wsh);
  k_update<<<(size_t)Bq * Nq * 16 / 256, 256, 0, stream>>>(
      memory, wsh, scal, e_arr, a_arr, out_w, out_mem);
}